// SI_LIF_23776938951033
// MI455X (gfx1250) — compile-verified
//
#include <hip/hip_runtime.h>

// ---------------------------------------------------------------------------
// SI-LIF ternary-spike scan for gfx1250 (MI455X).
// Pure streaming problem (256 MiB traffic, ~11us HBM floor @ 23.3 TB/s).
// Parallelism = B*dim = 8192 channels = 256 wave32 waves; the t-scan is
// serial, so each wave uses the CDNA5 async global->LDS DMA path
// (global_load_async_to_lds_b128 / s_wait_asynccnt) to triple-buffer its
// input stream while the VALU chain runs.
// ---------------------------------------------------------------------------

#define AS1 __attribute__((address_space(1)))
#define AS3 __attribute__((address_space(3)))

typedef int v4i __attribute__((vector_size(16)));

namespace {
constexpr int kB      = 8;
constexpr int kL      = 4096;
constexpr int kDim    = 1024;
constexpr int kLanes  = 32;                 // wave32; one wave per block
constexpr int kChunkT = 64;                 // timesteps per LDS chunk
constexpr int kNBuf   = 3;                  // chunks in flight (3*16=48 <= 63)
constexpr int kNChunk = kL / kChunkT;       // 64
constexpr int kChunkElems = kChunkT * kLanes;
}

#if __has_builtin(__builtin_amdgcn_s_wait_asynccnt)
#define WAIT_ASYNC(n) __builtin_amdgcn_s_wait_asynccnt(n)
#else
#define WAIT_ASYNC(n) asm volatile("s_wait_asynccnt %0" ::"i"(n) : "memory")
#endif

__device__ __forceinline__ void async_copy_b128(const float* g, float* l) {
#if __has_builtin(__builtin_amdgcn_global_load_async_to_lds_b128)
  __builtin_amdgcn_global_load_async_to_lds_b128(
      (AS1 v4i*)g, (AS3 v4i*)l, /*offset=*/0, /*cpol=*/0);
#else
  unsigned ldsOff = (unsigned)(size_t)(AS3 void*)l;
  unsigned long long ga = (unsigned long long)(size_t)g;
  asm volatile("global_load_async_to_lds_b128 %0, %1, off"
               ::"v"(ldsOff), "v"(ga)
               : "memory");
#endif
}

__global__ __launch_bounds__(kLanes) void silif_scan_kernel(
    const float* __restrict__ x,          // [B, L, dim]
    const float* __restrict__ phase,      // [B, 1]
    const float* __restrict__ beta_raw,   // [dim]
    const float* __restrict__ theta,      // [dim]
    const float* __restrict__ cstrength,  // [1]
    float* __restrict__ out)              // spikes [B,L,dim] ++ mem [B,dim]
{
  __shared__ float sm[kNBuf * kChunkElems];

  const int lane = threadIdx.x;                 // 0..31
  const int blk  = blockIdx.x;                  // 0..255
  const int b    = blk >> 5;                    // dim/32 = 32 tiles per batch
  const int d0   = (blk & 31) * kLanes;
  const int d    = d0 + lane;

  // beta = clip(sigmoid(beta_raw) + cs*phase, 0.01, 0.99)
  float beta = 1.0f / (1.0f + expf(-beta_raw[d]));
  beta += cstrength[0] * phase[b];
  beta = fminf(fmaxf(beta, 0.01f), 0.99f);
  const float th = theta[d];

  // Per-lane async-copy geometry: each b128 op moves 4 rows (timesteps) of
  // 32 floats; lane -> (row sub-index, 4-float column slice).
  const int rb = lane >> 3;           // 0..3 : row within group of 4
  const int cb = (lane & 7) * 4;      // float column offset (16B slice)
  const float* xlane = x + (size_t)b * kL * kDim + (size_t)(d0 + cb);

  auto issue_chunk = [&](int k) {
    float* lbuf = &sm[(k % kNBuf) * kChunkElems];
    const int t0 = k * kChunkT;
#pragma unroll
    for (int j = 0; j < kChunkT / 4; ++j) {
      const float* g = xlane + (size_t)(t0 + 4 * j + rb) * (size_t)kDim;
      float* l = lbuf + (4 * j + rb) * kLanes + cb;
      async_copy_b128(g, l);
    }
  };

  // Prologue: fill the pipeline.
  issue_chunk(0);
  issue_chunk(1);
  issue_chunk(2);

  float mem = 0.0f;
  float* outp = out + (size_t)b * kL * kDim + d;  // advances by kDim per t

#pragma unroll 1
  for (int k = 0; k < kNChunk; ++k) {
    // In-order completion: wait until only the chunks beyond k remain.
    if (k + 2 < kNChunk)      { WAIT_ASYNC(32); }
    else if (k + 1 < kNChunk) { WAIT_ASYNC(16); }
    else                      { WAIT_ASYNC(0);  }
    asm volatile("" ::: "memory");  // keep LDS reads below the wait

    const float* lbuf = &sm[(k % kNBuf) * kChunkElems];
#pragma unroll 8
    for (int t = 0; t < kChunkT; ++t) {
      float xv = lbuf[t * kLanes + lane];
      mem = fmaf(beta, mem, xv);                 // leak + integrate
      float sp = rintf(mem);                     // round-half-even == jnp.round
      sp = fminf(fmaxf(sp, -1.0f), 1.0f);        // ternary clip
      mem = fmaf(-sp, th, mem);                  // reset
      __builtin_nontemporal_store(sp, outp);     // streaming store (NT)
      outp += kDim;
    }
    asm volatile("" ::: "memory");  // buffer fully consumed before refill

    if (k + kNBuf < kNChunk) issue_chunk(k + kNBuf);
  }

  // Final membrane state, appended after spikes.
  out[(size_t)kB * kL * kDim + (size_t)b * kDim + d] = mem;
}

extern "C" void kernel_launch(void* const* d_in, const int* in_sizes, int n_in,
                              void* d_out, int out_size, void* d_ws,
                              size_t ws_size, hipStream_t stream) {
  (void)in_sizes; (void)n_in; (void)out_size; (void)d_ws; (void)ws_size;
  const float* x         = (const float*)d_in[0];
  const float* phase     = (const float*)d_in[1];
  const float* beta_raw  = (const float*)d_in[2];
  const float* theta     = (const float*)d_in[3];
  const float* cstrength = (const float*)d_in[4];
  float* out             = (float*)d_out;

  dim3 grid(kB * (kDim / kLanes));  // 256 blocks
  dim3 block(kLanes);               // 32 threads = 1 wave32
  silif_scan_kernel<<<grid, block, 0, stream>>>(x, phase, beta_raw, theta,
                                                cstrength, out);
}